// TreeLoss_10694468567626
// MI455X (gfx1250) — compile-verified
//
#include <hip/hip_runtime.h>
#include <stdint.h>

#define N_BODY 7
#define N_ACTION 52

constexpr int TILE = 32;            // batches per wave tile (== wave32)
constexpr int ACT_STRIDE = 60;      // floats; 240B rows -> 16B aligned, only 2-way LDS bank conflicts
constexpr int BODY_STRIDE = 8;      // floats
constexpr int WAVES_PER_BLOCK = 8;  // 256 threads

// fine-action -> coarse-body group boundaries (from _action2body)
__device__ __constant__ const int GSTART[7] = {0, 5, 11, 24, 32, 38, 48};
__device__ __constant__ const int GEND[7]   = {5, 11, 24, 32, 38, 48, 52};

__device__ __forceinline__ float fast_exp(float x) {
#if __has_builtin(__builtin_amdgcn_exp2f)
    return __builtin_amdgcn_exp2f(x * 1.4426950408889634f);   // v_exp_f32
#else
    return __expf(x);
#endif
}

__device__ __forceinline__ float fast_log(float x) {
#if __has_builtin(__builtin_amdgcn_logf)
    return __builtin_amdgcn_logf(x) * 0.6931471805599453f;    // v_log_f32
#else
    return __logf(x);
#endif
}

__device__ __forceinline__ float fast_sigmoid(float x) {
#if __has_builtin(__builtin_amdgcn_tanhf)
    // sigmoid(x) = 0.5*tanh(x/2) + 0.5 : single v_tanh_f32 (gfx1250)
    return __builtin_amdgcn_tanhf(x * 0.5f) * 0.5f + 0.5f;
#else
    return __builtin_amdgcn_rcpf(1.0f + fast_exp(-x));
#endif
}

// Low 32 bits of a generic pointer to a __shared__ object == LDS byte offset
__device__ __forceinline__ uint32_t lds_addr_of(const void* p) {
    return (uint32_t)(uintptr_t)p;
}

__global__ void __launch_bounds__(256) tree_loss_main(
    const float* __restrict__ body,     // [B,7]
    const float* __restrict__ act,      // [B,52]
    const int*   __restrict__ labA,     // [B]
    float*       __restrict__ wavePartials,
    int nTiles)
{
    __shared__ __align__(16) float sAct[WAVES_PER_BLOCK][TILE * ACT_STRIDE];
    __shared__ __align__(16) float sBody[WAVES_PER_BLOCK][TILE * BODY_STRIDE];

    const int lane  = threadIdx.x & 31;
    const int wib   = threadIdx.x >> 5;
    const int gwave = (int)((blockIdx.x * blockDim.x + threadIdx.x) >> 5);
    const int nWaves = (int)((gridDim.x * blockDim.x) >> 5);

    const uint32_t ldsActBase  = lds_addr_of(&sAct[wib][0]);
    const uint32_t ldsBodyBase = lds_addr_of(&sBody[wib][0]);
    const uint64_t actBase  = (uint64_t)(uintptr_t)act;
    const uint64_t bodyBase = (uint64_t)(uintptr_t)body;

    float acc = 0.0f;

    for (int tile = gwave; tile < nTiles; tile += nWaves) {
        const int base = tile * TILE;

        // Ensure previous iteration's LDS reads have completed before the
        // async engine overwrites the tile (async vs DS ordering is not
        // guaranteed by HW).
        asm volatile("s_wait_dscnt 0x0" ::: "memory");

        // --- stage action tile: 32 rows x 52 f32 = 416 x b128 chunks,
        //     consecutive chunks across lanes -> 512B contiguous / instr.
        //     Stream is read-once (252MB > 192MB L2): non-temporal hint. ---
        {
            const uint32_t g0 = (uint32_t)base * (uint32_t)(N_ACTION * 4);
            #pragma unroll
            for (int k = 0; k < 13; ++k) {
                const int chunk = k * 32 + lane;
                const int row = chunk / 13;          // magic-mul, cheap VALU
                const int col = chunk - row * 13;    // float4 column
                const uint32_t loff = ldsActBase +
                    (uint32_t)(row * ACT_STRIDE + col * 4) * 4u;
                const uint32_t goff = g0 + (uint32_t)chunk * 16u;
                asm volatile("global_load_async_to_lds_b128 %0, %1, %2 th:TH_LOAD_NT"
                             :: "v"(loff), "v"(goff), "s"(actBase) : "memory");
            }
        }
        // --- stage body tile: 32 rows x 7 f32 = 224 x b32 elems ---
        {
            const uint32_t g0 = (uint32_t)base * (uint32_t)(N_BODY * 4);
            #pragma unroll
            for (int k = 0; k < 7; ++k) {
                const int e   = k * 32 + lane;
                const int row = e / 7;
                const int col = e - row * 7;
                const uint32_t loff = ldsBodyBase +
                    (uint32_t)(row * BODY_STRIDE + col) * 4u;
                const uint32_t goff = g0 + (uint32_t)e * 4u;
                asm volatile("global_load_async_to_lds_b32 %0, %1, %2 th:TH_LOAD_NT"
                             :: "v"(loff), "v"(goff), "s"(bodyBase) : "memory");
            }
        }

        // overlap: label load (LOADcnt) while async copies run
        const int la = __builtin_nontemporal_load(labA + base + lane);

        asm volatile("s_wait_asynccnt 0x0" ::: "memory");

        // --- gather my row from LDS (row per lane, 16B-aligned) ---
        const float* myAct  = &sAct[wib][lane * ACT_STRIDE];
        const float* myBody = &sBody[wib][lane * BODY_STRIDE];

        float a[N_ACTION];
        #pragma unroll
        for (int k = 0; k < 13; ++k) {
            const float4 v = *reinterpret_cast<const float4*>(myAct + k * 4);
            a[k * 4 + 0] = v.x; a[k * 4 + 1] = v.y;
            a[k * 4 + 2] = v.z; a[k * 4 + 3] = v.w;
        }
        float b[N_BODY];
        #pragma unroll
        for (int i = 0; i < N_BODY; ++i) b[i] = myBody[i];

        // coarse group of the label
        const int gla = (la <= 4)  ? 0 :
                        (la <= 10) ? 1 :
                        (la <= 23) ? 2 :
                        (la <= 31) ? 3 :
                        (la <= 37) ? 4 :
                        (la <= 47) ? 5 : 6;

        float z = 0.0f, sa_la = 0.0f, sb_la = 0.0f;
        #pragma unroll
        for (int g = 0; g < 7; ++g) {
            float E = 0.0f;
            #pragma unroll
            for (int aa = GSTART[g]; aa < GEND[g]; ++aa) {
                const float s = fast_sigmoid(a[aa]);
                if (aa == la) sa_la = s;
                E += fast_exp(s);
            }
            const float sbg = fast_sigmoid(b[g]);
            if (g == gla) sb_la = sbg;
            z += fast_exp(sbg) * (1.0f + E);
        }
        acc += fast_log(z) - sa_la - sb_la;   // -log(marginal/z)
    }

    // deterministic wave32 reduction
    #pragma unroll
    for (int off = 16; off > 0; off >>= 1)
        acc += __shfl_xor(acc, off, 32);
    if (lane == 0) wavePartials[gwave] = acc;
}

__global__ void __launch_bounds__(256) tree_loss_reduce(
    const float* __restrict__ partials, int n, float invB,
    float* __restrict__ out)
{
    __shared__ float sw[WAVES_PER_BLOCK];
    float acc = 0.0f;
    for (int i = (int)threadIdx.x; i < n; i += (int)blockDim.x)
        acc += partials[i];
    #pragma unroll
    for (int off = 16; off > 0; off >>= 1)
        acc += __shfl_xor(acc, off, 32);
    const int lane = threadIdx.x & 31;
    const int w    = threadIdx.x >> 5;
    if (lane == 0) sw[w] = acc;
    __syncthreads();
    if (threadIdx.x == 0) {
        float t = 0.0f;
        #pragma unroll
        for (int i = 0; i < WAVES_PER_BLOCK; ++i) t += sw[i];
        out[0] = t * invB;
    }
}

extern "C" void kernel_launch(void* const* d_in, const int* in_sizes, int n_in,
                              void* d_out, int out_size, void* d_ws, size_t ws_size,
                              hipStream_t stream) {
    const float* body = (const float*)d_in[0];   // [B,7]
    const float* act  = (const float*)d_in[1];   // [B,52]
    // d_in[2] = labels_body (unused by the reference loss)
    const int*   labA = (const int*)d_in[3];     // [B]
    float* outF = (float*)d_out;

    const int B      = in_sizes[0] / N_BODY;     // 1048576
    const int nTiles = B / TILE;                 // 32768

    const int blocks = 1024;                     // 8192 waves, 4 tiles/wave
    const int nWaves = blocks * WAVES_PER_BLOCK;
    float* partials = (float*)d_ws;              // nWaves floats (32 KB)

    tree_loss_main<<<blocks, 256, 0, stream>>>(body, act, labA, partials, nTiles);
    tree_loss_reduce<<<1, 256, 0, stream>>>(partials, nWaves, 1.0f / (float)B, outF);
}